// OPTAttention_29085518528997
// MI455X (gfx1250) — compile-verified
//
#include <hip/hip_runtime.h>
#include <hip/hip_bf16.h>
#include <math.h>

// ---------------------------------------------------------------------------
// OPT-1.3B decode attention w/ paged KV cache, for gfx1250 (MI455X, wave32).
//   S=64 seqs, E=2048, H=32, D=64, L=2048, BS=16, BPS=128
// Pipeline: [qkv = hs@wqkv+b] -> [paged flash attention] -> [out = attn@wout+b]
// GEMMs use V_WMMA_F32_16X16X4_F32; attention is pure HBM streaming
// (~1.6-2.1 GB of KV @ 23.3 TB/s dominates at ~70-90 us; projections ~3 us).
// ---------------------------------------------------------------------------

typedef float v2f __attribute__((ext_vector_type(2)));
typedef float v8f __attribute__((ext_vector_type(8)));

#define SEQS   64
#define EMB    2048
#define HEADS  32
#define HDIM   64
#define CTXMAX 2048
#define BLKSZ  16
#define BPS    128

// ---------------------------------------------------------------------------
// GEMM: C[64 x N] = A[64 x K] @ B[K x N] + bias[N]      (fp32, WMMA f32 16x16x4)
// One block per 16-wide N tile; 4 waves split K, LDS-reduce partial C tiles.
// A (512KB) is L2-resident; B streams from HBM exactly once.
// N, K are template constants so every fragment load folds into a 24-bit
// immediate offset off a single global base pointer (A: mt*16*K*4 <= 393KB,
// B: (4u+1)*N*4 <= 320KB < 8MB imm) -> pure global_load path, 16 v_wmma and
// 2 pointer bumps per 16-K iteration.
// ---------------------------------------------------------------------------
template <int N, int K>
__global__ __launch_bounds__(128)
void gemm64_wmma_f32(const float* __restrict__ A, const float* __restrict__ B,
                     const float* __restrict__ bias, float* __restrict__ C) {
    const int n0   = blockIdx.x * 16;
    const int tid  = threadIdx.x;
    const int wave = tid >> 5;
    const int lane = tid & 31;
    const int half = lane >> 4;     // 0: lanes 0-15, 1: lanes 16-31
    const int r    = lane & 15;

    __shared__ float red[4 * 32 * 32];   // [wave][lane][4 mtiles * 8 vgprs]

    v8f acc[4];
#pragma unroll
    for (int mt = 0; mt < 4; ++mt)
#pragma unroll
        for (int i = 0; i < 8; ++i) acc[mt][i] = 0.0f;

    // Each wave owns a contiguous quarter of K.
    constexpr int kPerWave = K >> 2;
    const int kbeg = wave * kPerWave;

    // Single loop-carried base pointer per operand; all fragment element
    // addresses are compile-time immediate offsets (ISA §7.12.2 layouts:
    //  A frag 16x4 : lane r holds row, VGPR.x/.y = K = kbase+2*half (+1)
    //  B frag 4x16 : lane r holds col n0+r, same K striping as A).
    const float* ap = A + (size_t)r * K + kbeg + 2 * half;
    const float* bp = B + ((size_t)kbeg + 2 * half) * N + n0 + r;

    for (int ks = 0; ks < kPerWave; ks += 16) {
        v2f bf[4];
#pragma unroll
        for (int u = 0; u < 4; ++u) {
            bf[u].x = bp[(4 * u)     * N];
            bf[u].y = bp[(4 * u + 1) * N];
        }
        v2f af[4][4];
#pragma unroll
        for (int u = 0; u < 4; ++u)
#pragma unroll
            for (int mt = 0; mt < 4; ++mt) {
                af[u][mt].x = ap[mt * 16 * K + 4 * u];
                af[u][mt].y = ap[mt * 16 * K + 4 * u + 1];
            }
#pragma unroll
        for (int u = 0; u < 4; ++u)
#pragma unroll
            for (int mt = 0; mt < 4; ++mt)
                acc[mt] = __builtin_amdgcn_wmma_f32_16x16x4_f32(
                    false, af[u][mt], false, bf[u], (short)0, acc[mt],
                    false, false);
        ap += 16;
        bp += (size_t)16 * N;
    }

    // Spill partial tiles, reduce across the 4 K-split waves.
#pragma unroll
    for (int mt = 0; mt < 4; ++mt)
#pragma unroll
        for (int i = 0; i < 8; ++i)
            red[(wave * 32 + lane) * 32 + mt * 8 + i] = acc[mt][i];
    __syncthreads();

    // 128 threads x 8 outputs = 1024 = 64 rows x 16 cols.
    const int jbase = (tid >> 5) * 8;
#pragma unroll
    for (int jj = 0; jj < 8; ++jj) {
        const int j = jbase + jj;
        float s = red[(0 * 32 + lane) * 32 + j]
                + red[(1 * 32 + lane) * 32 + j]
                + red[(2 * 32 + lane) * 32 + j]
                + red[(3 * 32 + lane) * 32 + j];
        const int mt = j >> 3, i = j & 7;
        const int m = mt * 16 + i + 8 * half;     // C/D layout: M = vgpr + 8*half
        const int n = n0 + r;
        C[(size_t)m * N + n] = s + bias[n];
    }
}

// ---------------------------------------------------------------------------
// Paged decode attention. One workgroup per (head, seq), 256 threads.
// Softmax over ctx cached positions + the freshly projected k/v token.
// Bandwidth-bound: K pass is lane-per-position float4 streaming (256B-aligned
// rows), V pass is 64-lane-per-row fully coalesced.
// ---------------------------------------------------------------------------
__global__ __launch_bounds__(256)
void paged_attn_f32(const float* __restrict__ qkv,
                    const float* __restrict__ key_cache,
                    const float* __restrict__ value_cache,
                    const int*   __restrict__ block_tables,
                    const int*   __restrict__ context_lens,
                    float* __restrict__ attn_out) {
    const int h   = blockIdx.x;      // head
    const int s   = blockIdx.y;      // sequence
    const int tid = threadIdx.x;

    __shared__ float q_sh[HDIM];
    __shared__ int   btab[BPS];
    __shared__ float scores[CTXMAX + 16];
    __shared__ float redbuf[256];
    __shared__ float vpart[4 * HDIM];

    const int ctx = context_lens[s];
    const int n   = ctx + 1;                       // + current token

    const float* qkv_s = qkv + (size_t)s * (3 * EMB);
    if (tid < HDIM) q_sh[tid] = qkv_s[h * HDIM + tid];
    if (tid < BPS)  btab[tid] = block_tables[s * BPS + tid];
    __syncthreads();

    // ---- scores = (q . K) * scale -----------------------------------------
    const float scale = 0.125f;                    // 1/sqrt(64)
    float lmax = -3.0e38f;
    for (int l = tid; l < n; l += 256) {
        const float* kp;
        if (l < ctx) {
            const int blk = btab[l >> 4];
            kp = key_cache +
                 (((size_t)blk * BLKSZ + (l & 15)) * HEADS + h) * HDIM;
        } else {
            kp = qkv_s + EMB + h * HDIM;           // k_new from QKV projection
        }
        // prefetch next iteration's K row (gfx1250 global_prefetch path)
        if (l + 256 < ctx) {
            const int nblk = btab[(l + 256) >> 4];
            __builtin_prefetch(key_cache +
                (((size_t)nblk * BLKSZ + ((l + 256) & 15)) * HEADS + h) * HDIM,
                0, 1);
        }
        float acc = 0.0f;
#pragma unroll
        for (int d4 = 0; d4 < HDIM / 4; ++d4) {
            const float4 k4 = ((const float4*)kp)[d4];
            acc += k4.x * q_sh[d4 * 4 + 0];
            acc += k4.y * q_sh[d4 * 4 + 1];
            acc += k4.z * q_sh[d4 * 4 + 2];
            acc += k4.w * q_sh[d4 * 4 + 3];
        }
        acc *= scale;
        scores[l] = acc;
        lmax = fmaxf(lmax, acc);
    }

    // ---- block max ---------------------------------------------------------
    redbuf[tid] = lmax;
    __syncthreads();
    for (int off = 128; off > 0; off >>= 1) {
        if (tid < off) redbuf[tid] = fmaxf(redbuf[tid], redbuf[tid + off]);
        __syncthreads();
    }
    const float m = redbuf[0];
    __syncthreads();

    // ---- exp + block sum ---------------------------------------------------
    float lsum = 0.0f;
    for (int l = tid; l < n; l += 256) {
        const float e = __expf(scores[l] - m);
        scores[l] = e;
        lsum += e;
    }
    redbuf[tid] = lsum;
    __syncthreads();
    for (int off = 128; off > 0; off >>= 1) {
        if (tid < off) redbuf[tid] += redbuf[tid + off];
        __syncthreads();
    }
    const float inv = 1.0f / redbuf[0];

    // ---- out = probs @ V ---------------------------------------------------
    const int dd = tid & (HDIM - 1);               // 0..63
    const int g  = tid >> 6;                       // 0..3 position groups
    float acc = 0.0f;
    for (int l = g; l < n; l += 4) {
        float v;
        if (l < ctx) {
            const int blk = btab[l >> 4];
            v = value_cache[(((size_t)blk * BLKSZ + (l & 15)) * HEADS + h) * HDIM + dd];
        } else {
            v = qkv_s[2 * EMB + h * HDIM + dd];    // v_new
        }
        acc += scores[l] * v;
    }
    vpart[g * HDIM + dd] = acc;
    __syncthreads();
    if (tid < HDIM) {
        const float o = (vpart[tid] + vpart[HDIM + tid] +
                         vpart[2 * HDIM + tid] + vpart[3 * HDIM + tid]) * inv;
        attn_out[(size_t)s * EMB + h * HDIM + tid] = o;
    }
}

// ---------------------------------------------------------------------------
extern "C" void kernel_launch(void* const* d_in, const int* in_sizes, int n_in,
                              void* d_out, int out_size, void* d_ws, size_t ws_size,
                              hipStream_t stream) {
    const float* hidden = (const float*)d_in[0];   // [64, 2048]
    const float* wqkv   = (const float*)d_in[1];   // [2048, 6144]
    const float* bqkv   = (const float*)d_in[2];   // [6144]
    const float* wout   = (const float*)d_in[3];   // [2048, 2048]
    const float* bout   = (const float*)d_in[4];   // [2048]
    const float* kcache = (const float*)d_in[5];   // [8192, 16, 32, 64]
    const float* vcache = (const float*)d_in[6];   // [8192, 16, 32, 64]
    const int*   btab   = (const int*)d_in[7];     // [64, 128]
    const int*   clens  = (const int*)d_in[8];     // [64]
    float*       out    = (float*)d_out;           // [64, 2048]

    float* qkv  = (float*)d_ws;                    // [64, 6144]  (1.5 MB)
    float* attn = qkv + (size_t)SEQS * 3 * EMB;    // [64, 2048]  (0.5 MB)

    // 1) fused QKV projection: qkv = hidden @ wqkv + bqkv
    gemm64_wmma_f32<3 * EMB, EMB><<<dim3(3 * EMB / 16), 128, 0, stream>>>(
        hidden, wqkv, bqkv, qkv);

    // 2) paged decode attention (HBM-bound: ~1.6-2.1 GB of KV @ 23.3 TB/s)
    paged_attn_f32<<<dim3(HEADS, SEQS), 256, 0, stream>>>(
        qkv, kcache, vcache, btab, clens, attn);

    // 3) output projection: out = attn @ wout + bout
    gemm64_wmma_f32<EMB, EMB><<<dim3(EMB / 16), 128, 0, stream>>>(
        attn, wout, bout, out);
}